// NetDensity_4647154614487
// MI455X (gfx1250) — compile-verified
//
#include <hip/hip_runtime.h>

// ---------------------------------------------------------------------------
// Problem constants (match reference)
// ---------------------------------------------------------------------------
#define NBX 256
#define NBY 256
#define XL_ 0.0f
#define YL_ 0.0f
#define BSX_ 2.0f   // (512-0)/256
#define BSY_ 2.0f

#define WAVES_PER_BLOCK 8
#define CHUNK 64            // nets staged into LDS per wave per stage

typedef __attribute__((ext_vector_type(2))) float v2f;
typedef __attribute__((ext_vector_type(8))) float v8f;

__constant__ float RISA_TAB[47] = {
    1.0f, 1.0f, 1.0f, 1.0f,
    1.0828f, 1.1536f, 1.2206f, 1.2823f, 1.3385f, 1.3991f, 1.4493f,
    1.6899f, 1.6899f, 1.6899f, 1.6899f, 1.6899f,
    1.8924f, 1.8924f, 1.8924f, 1.8924f, 1.8924f,
    2.0743f, 2.0743f, 2.0743f, 2.0743f, 2.0743f,
    2.2334f, 2.2334f, 2.2334f, 2.2334f, 2.2334f,
    2.3892f, 2.3892f, 2.3892f, 2.3892f, 2.3892f,
    2.5356f, 2.5356f, 2.5356f, 2.5356f, 2.5356f,
    2.6625f, 2.6625f, 2.6625f, 2.6625f, 2.6625f,
    2.7933f
};

// ---------------------------------------------------------------------------
// Pass 1: per-net bounding boxes + RISA-weighted demand coefficients (SoA)
// ---------------------------------------------------------------------------
__global__ __launch_bounds__(256) void bbox_kernel(
    const float* __restrict__ pin_pos,
    const int*   __restrict__ netpin_start,
    const int*   __restrict__ flat_netpin,
    const float* __restrict__ net_weights,
    float* __restrict__ xmin_o, float* __restrict__ xmax_o,
    float* __restrict__ ymin_o, float* __restrict__ ymax_o,
    float* __restrict__ ch_o,   float* __restrict__ cv_o,
    int num_nets)
{
    int n = blockIdx.x * blockDim.x + threadIdx.x;
    if (n >= num_nets) return;
    int s = netpin_start[n];
    int e = netpin_start[n + 1];
    float xmn =  3.4e38f, xmx = -3.4e38f;
    float ymn =  3.4e38f, ymx = -3.4e38f;
    for (int i = s; i < e; ++i) {
        int p = flat_netpin[i];
        float x = pin_pos[2 * p];
        float y = pin_pos[2 * p + 1];
        xmn = fminf(xmn, x); xmx = fmaxf(xmx, x);
        ymn = fminf(ymn, y); ymx = fmaxf(ymx, y);
    }
    int cnt = e - s;
    if (cnt > 46) cnt = 46;
    float wt = RISA_TAB[cnt] * net_weights[n];
    float dx = xmx - xmn;
    float dy = ymx - ymn;
    float ch = (dy > 0.0f) ? wt / fmaxf(dy, 1e-12f) : 0.0f;
    float cv = (dx > 0.0f) ? wt / fmaxf(dx, 1e-12f) : 0.0f;
    xmin_o[n] = xmn; xmax_o[n] = xmx;
    ymin_o[n] = ymn; ymax_o[n] = ymx;
    ch_o[n] = ch;    cv_o[n] = cv;
}

// ---------------------------------------------------------------------------
// Pass 2: H/V density as a K=num_nets contraction via v_wmma_f32_16x16x4_f32.
// One block per 16x16 output tile; 8 waves split K; deterministic LDS reduce.
//
// A-fragment layout (16x4 f32, wave32): lane<16 holds M=lane, K={0,1};
// lane>=16 holds M=lane-16, K={2,3}.  B (4x16): VGPR j holds K=2*j+half.
// C/D (16x16 f32): VGPR j -> (M=j, N=lane) lanes 0-15, (M=8+j, N=lane-16).
// ---------------------------------------------------------------------------
__global__ __launch_bounds__(256) void density_wmma_kernel(
    const float* __restrict__ xmin_i, const float* __restrict__ xmax_i,
    const float* __restrict__ ymin_i, const float* __restrict__ ymax_i,
    const float* __restrict__ ch_i,   const float* __restrict__ cv_i,
    float* __restrict__ out,
    int num_nets)
{
    __shared__ float sA[WAVES_PER_BLOCK][6][CHUNK];   // xmin,xmax,ch,cv,ymin,ymax
    __shared__ float redH[WAVES_PER_BLOCK][256];
    __shared__ float redV[WAVES_PER_BLOCK][256];

    const int tid  = threadIdx.x;
    const int w    = tid >> 5;        // wave id within block
    const int lane = tid & 31;
    const int half = lane >> 4;       // 0: K pair {0,1}; 1: K pair {2,3}
    const int m    = lane & 15;       // M (A) / N (B) index within tile

    const int bx = blockIdx.x;        // x-tile
    const int by = blockIdx.y;        // y-tile

    const float bxl = XL_ + BSX_ * (float)(bx * 16 + m);
    const float bxr = bxl + BSX_;
    const float byl = YL_ + BSY_ * (float)(by * 16 + m);
    const float byr = byl + BSY_;

    v8f accH = {};
    v8f accV = {};

    const int kstripe = num_nets / WAVES_PER_BLOCK;   // 32768, multiple of CHUNK
    const int kbase0  = w * kstripe;

    for (int c = 0; c < kstripe; c += CHUNK) {
        // --- stage CHUNK nets into this wave's LDS slab (coalesced) ---
        int g0 = kbase0 + c + lane;
        int g1 = g0 + 32;
        sA[w][0][lane] = xmin_i[g0];  sA[w][0][lane + 32] = xmin_i[g1];
        sA[w][1][lane] = xmax_i[g0];  sA[w][1][lane + 32] = xmax_i[g1];
        sA[w][2][lane] = ch_i[g0];    sA[w][2][lane + 32] = ch_i[g1];
        sA[w][3][lane] = cv_i[g0];    sA[w][3][lane + 32] = cv_i[g1];
        sA[w][4][lane] = ymin_i[g0];  sA[w][4][lane + 32] = ymin_i[g1];
        sA[w][5][lane] = ymax_i[g0];  sA[w][5][lane + 32] = ymax_i[g1];
        __syncthreads();

        #pragma unroll 4
        for (int k = 0; k < CHUNK; k += 4) {
            // A side: this lane covers nets k+2*half and k+2*half+1
            int iA0 = k + 2 * half;
            int iA1 = iA0 + 1;
            float ox0 = fmaxf(fminf(sA[w][1][iA0], bxr) - fmaxf(sA[w][0][iA0], bxl), 0.0f);
            float ox1 = fmaxf(fminf(sA[w][1][iA1], bxr) - fmaxf(sA[w][0][iA1], bxl), 0.0f);
            v2f aH = { sA[w][2][iA0] * ox0, sA[w][2][iA1] * ox1 };
            v2f aV = { sA[w][3][iA0] * ox0, sA[w][3][iA1] * ox1 };

            // B side: VGPR j holds K = 2*j + half  -> nets k+half, k+2+half
            int iB0 = k + half;
            int iB1 = k + 2 + half;
            float oy0 = fmaxf(fminf(sA[w][5][iB0], byr) - fmaxf(sA[w][4][iB0], byl), 0.0f);
            float oy1 = fmaxf(fminf(sA[w][5][iB1], byr) - fmaxf(sA[w][4][iB1], byl), 0.0f);
            v2f b = { oy0, oy1 };

            // D = A x B + C, full f32 tensor path
            accH = __builtin_amdgcn_wmma_f32_16x16x4_f32(
                false, aH, false, b, (short)0, accH, false, false);
            accV = __builtin_amdgcn_wmma_f32_16x16x4_f32(
                false, aV, false, b, (short)0, accV, false, false);
        }
        __syncthreads();
    }

    // --- dump per-wave partial tiles to LDS in canonical (M,N) order ---
    #pragma unroll
    for (int j = 0; j < 8; ++j) {
        int Mrow = half ? (8 + j) : j;
        int idx  = Mrow * 16 + m;
        redH[w][idx] = accH[j];
        redV[w][idx] = accV[j];
    }
    __syncthreads();

    // --- fixed-order cross-wave reduction (deterministic), then store ---
    float h = 0.0f, v = 0.0f;
    #pragma unroll
    for (int ww = 0; ww < WAVES_PER_BLOCK; ++ww) {
        h += redH[ww][tid];
        v += redV[ww][tid];
    }
    int X = bx * 16 + (tid >> 4);
    int Y = by * 16 + (tid & 15);
    int o = X * NBY + Y;
    out[o]                 = fabsf(h) + fabsf(v);   // |H| + |V|
    out[NBX * NBY + o]     = h;                     // H
    out[2 * NBX * NBY + o] = v;                     // V
}

// ---------------------------------------------------------------------------
// Host-side launcher
// ---------------------------------------------------------------------------
extern "C" void kernel_launch(void* const* d_in, const int* in_sizes, int n_in,
                              void* d_out, int out_size, void* d_ws, size_t ws_size,
                              hipStream_t stream) {
    const float* pin_pos      = (const float*)d_in[0];
    const int*   netpin_start = (const int*)  d_in[1];
    const int*   flat_netpin  = (const int*)  d_in[2];
    const float* net_weights  = (const float*)d_in[3];
    float*       out          = (float*)d_out;

    const int num_nets = in_sizes[1] - 1;   // 262144

    // SoA scratch: 6 arrays of num_nets floats (~6.3 MB)
    float* ws   = (float*)d_ws;
    float* xmin = ws + 0 * (size_t)num_nets;
    float* xmax = ws + 1 * (size_t)num_nets;
    float* ymin = ws + 2 * (size_t)num_nets;
    float* ymax = ws + 3 * (size_t)num_nets;
    float* ch   = ws + 4 * (size_t)num_nets;
    float* cv   = ws + 5 * (size_t)num_nets;

    int blocks1 = (num_nets + 255) / 256;
    bbox_kernel<<<blocks1, 256, 0, stream>>>(
        pin_pos, netpin_start, flat_netpin, net_weights,
        xmin, xmax, ymin, ymax, ch, cv, num_nets);

    dim3 grid2(NBX / 16, NBY / 16);   // 16 x 16 output tiles
    density_wmma_kernel<<<grid2, 256, 0, stream>>>(
        xmin, xmax, ymin, ymax, ch, cv, out, num_nets);
}